// CombinedModel_80315888435653
// MI455X (gfx1250) — compile-verified
//
#include <hip/hip_runtime.h>
#include <hip/hip_bf16.h>
#include <math.h>

#define N_E      50125
#define BANDW    300
#define NELEM    30
#define HID      150
#define HPAD     152
#define N_NEW    20000
#define N_CHAN   4096
#define KTILES   37        // 37*4 = 148 of K=150, remainder 2 handled scalar
#define EGRP     5         // element groups (deterministic split of the e-sum)
#define EPB      6         // elements per group (5*6 = 30)
#define NGRP32   ((N_E + 31) / 32)   // 1567 32-column wave groups

typedef __attribute__((ext_vector_type(2))) float v2f;
typedef __attribute__((ext_vector_type(8))) float v8f;

__device__ __forceinline__ float exp10_fast(float v) {
    // 10^v = 2^(v * log2(10)); maps to v_exp_f32
    return exp2f(v * 3.32192809488736235f);
}

// ---------------------------------------------------------------------------
// Kernel 1: per-element 3-layer MLP -> h3, one block per element.
// ---------------------------------------------------------------------------
__global__ void __launch_bounds__(160)
mlp_kernel(const float* __restrict__ temp,
           const float* __restrict__ W1, const float* __restrict__ b1,
           const float* __restrict__ W2, const float* __restrict__ b2,
           const float* __restrict__ W3, const float* __restrict__ b3,
           float* __restrict__ H) {
    __shared__ float sh0[HID];
    __shared__ float sh1[HID];
    const int e = blockIdx.x;
    const int k = threadIdx.x;
    const float t = temp[0];

    if (k < HID) sh0[k] = tanhf(t * W1[e * HID + k] + b1[e * HID + k]);
    __syncthreads();

    if (k < HID) {
        const float* __restrict__ W = W2 + (size_t)e * HID * HID;
        float acc = b2[e * HID + k];
        for (int h = 0; h < HID; ++h) acc = fmaf(sh0[h], W[h * HID + k], acc);
        sh1[k] = tanhf(acc);
    }
    __syncthreads();

    float h3 = 0.0f;
    if (k < HID) {
        const float* __restrict__ W = W3 + (size_t)e * HID * HID;
        float acc = b3[e * HID + k];
        for (int h = 0; h < HID; ++h) acc = fmaf(sh1[h], W[h * HID + k], acc);
        h3 = tanhf(acc);
    }
    if (k < HPAD) H[e * HPAD + k] = (k < HID) ? h3 : 0.0f;  // zero the pad
}

// ---------------------------------------------------------------------------
// Kernel 2: the 902 MB Wout GEMV via V_WMMA_F32_16X16X4_F32, fused with
// bout + 10^(o*scales+means) * abundance.  grid.y splits the 30-element
// reduction into 5 deterministic partial sums (ypart) for occupancy.
// One wave handles two 16-column tiles; branch-free A fragment (h in row
// M=0, built with a per-lane 0/1 mask); B addressed by incremented pointers.
// ---------------------------------------------------------------------------
__global__ void __launch_bounds__(256)
wout_y_kernel(const float* __restrict__ Wout, const float* __restrict__ bout,
              const float* __restrict__ scales, const float* __restrict__ means,
              const float* __restrict__ abund, const float* __restrict__ H,
              float* __restrict__ ypart) {
    __shared__ float hs[EPB * HPAD];
    const int tid  = threadIdx.x;
    const int egrp = blockIdx.y;
    const int e0   = egrp * EPB;
    for (int i = tid; i < EPB * HPAD; i += 256) hs[i] = H[e0 * HPAD + i];
    __syncthreads();

    const int lane = tid & 31;
    const int wave = tid >> 5;
    const int grp  = blockIdx.x * 8 + wave;        // 32-column group
    if (grp >= NGRP32) return;                     // wave-uniform exit

    const int   base = grp * 32;
    const int   sub  = lane & 15;
    const bool  loH  = (lane < 16);
    const int   koff = loH ? 0 : 2;                // A/B K-interleave per half
    const float am   = (sub == 0) ? 1.0f : 0.0f;   // only M=0 row of A is live

    const int col0 = base + sub;
    const int col1 = base + 16 + sub;
    const int cc0  = (col0 < N_E) ? col0 : (N_E - 1);  // clamp: dead lanes only
    const int cc1  = (col1 < N_E) ? col1 : (N_E - 1);
    const size_t off0 = (size_t)koff * N_E + cc0;
    const size_t off1 = (size_t)koff * N_E + cc1;

    float yacc0 = 0.0f, yacc1 = 0.0f;
    for (int el = 0; el < EPB; ++el) {
        const int e = e0 + el;
        const float* __restrict__ W   = Wout + (size_t)e * HID * N_E;
        const float* __restrict__ hp  = hs + el * HPAD + koff;
        const float* __restrict__ pw0 = W + off0;
        const float* __restrict__ pw1 = W + off1;

        v8f c0 = {0.f, 0.f, 0.f, 0.f, 0.f, 0.f, 0.f, 0.f};
        v8f c1 = {0.f, 0.f, 0.f, 0.f, 0.f, 0.f, 0.f, 0.f};

        for (int kt = 0; kt < KTILES; ++kt) {
            // A fragment: ds_load_b64 + 2 VALU, no divergent control flow
            const v2f hv = *(const v2f*)hp;        // h[k+koff], h[k+1+koff]
            v2f a;  a.x = hv.x * am;  a.y = hv.y * am;
            // B fragments: rows K=koff (reg0) and K=koff+1 (reg1), N across lanes
            v2f b0, b1;
            b0.x = pw0[0];  b0.y = pw0[N_E];
            b1.x = pw1[0];  b1.y = pw1[N_E];
            c0 = __builtin_amdgcn_wmma_f32_16x16x4_f32(
                     false, a, false, b0, (short)0, c0, false, false);
            c1 = __builtin_amdgcn_wmma_f32_16x16x4_f32(
                     false, a, false, b1, (short)0, c1, false, false);
            hp  += 4;
            pw0 += 4 * (size_t)N_E;
            pw1 += 4 * (size_t)N_E;
        }

        // K remainder (rows 148,149) + bias + 10^(o*s+m) epilogue
        const float h148 = hs[el * HPAD + 148];
        const float h149 = hs[el * HPAD + 149];
        const float* __restrict__ pr0 = W + (size_t)148 * N_E + cc0;
        const float* __restrict__ pr1 = W + (size_t)148 * N_E + cc1;
        float o0 = c0[0];                          // row M=0, N = sub (lanes 0-15)
        float o1 = c1[0];
        o0 = fmaf(h148, pr0[0], o0);  o0 = fmaf(h149, pr0[N_E], o0);
        o1 = fmaf(h148, pr1[0], o1);  o1 = fmaf(h149, pr1[N_E], o1);
        o0 += bout[(size_t)e * N_E + cc0];
        o1 += bout[(size_t)e * N_E + cc1];
        const float ab = (e < 5) ? 1.0f : abund[e];
        const float v0 = fmaf(o0, scales[(size_t)e * N_E + cc0],
                                  means[(size_t)e * N_E + cc0]);
        const float v1 = fmaf(o1, scales[(size_t)e * N_E + cc1],
                                  means[(size_t)e * N_E + cc1]);
        yacc0 += ab * exp10_fast(v0);
        yacc1 += ab * exp10_fast(v1);
    }

    float* __restrict__ yp = ypart + (size_t)egrp * N_E;
    if (loH) {
        if (col0 < N_E) yp[col0] = yacc0;
        if (col1 < N_E) yp[col1] = yacc1;
    }
}

// ---------------------------------------------------------------------------
// Kernel 2b: deterministic reduction of the 5 element-group partials.
// ---------------------------------------------------------------------------
__global__ void combine_y_kernel(const float* __restrict__ ypart,
                                 float* __restrict__ y) {
    const int n = blockIdx.x * blockDim.x + threadIdx.x;
    if (n >= N_E) return;
    float s = 0.0f;
    for (int g = 0; g < EGRP; ++g) s += ypart[(size_t)g * N_E + n];
    y[n] = s;
}

// ---------------------------------------------------------------------------
// Kernel 3: Gaussian velocity broadening, one thread per output sample.
// x/dx/y (~600 KB) stay resident in L2.
// ---------------------------------------------------------------------------
__global__ void broaden_kernel(const float* __restrict__ x, const float* __restrict__ dx,
                               const float* __restrict__ y, const float* __restrict__ vel,
                               float* __restrict__ broad) {
    const int i = blockIdx.x * blockDim.x + threadIdx.x;
    if (i >= N_E) return;
    const float v     = fmaxf(vel[0], 1e-30f);
    const float stdev = v * 1000.0f / 299792458.0f;
    const float xi    = x[i];
    const float inv   = 1.0f / (xi * stdev);
    const int j0 = (i - BANDW > 0) ? (i - BANDW) : 0;
    const int j1 = (i + BANDW < N_E - 1) ? (i + BANDW) : (N_E - 1);
    float num = 0.0f, den = 0.0f;
    for (int j = j0; j <= j1; ++j) {
        const float d = (x[j] - xi) * inv;
        const float w = __expf(-0.5f * d * d) * dx[j];
        den += w;
        num = fmaf(w, y[j], num);
    }
    broad[i] = num / den;
}

// ---------------------------------------------------------------------------
// Kernel 4: redshift + interp onto new_x + response fold.
// interp over ecent = x/(1+z)  <=>  binary search x for u = new_x*(1+z).
// ---------------------------------------------------------------------------
__global__ void interp_fold_kernel(const float* __restrict__ x, const float* __restrict__ broad,
                                   const float* __restrict__ new_x, const float* __restrict__ resp,
                                   const float* __restrict__ logz, const float* __restrict__ norm,
                                   float* __restrict__ folded) {
    const int m = blockIdx.x * blockDim.x + threadIdx.x;
    if (m >= N_NEW) return;
    const float z   = exp10_fast(logz[0]);
    const float zp1 = 1.0f + z;
    const float u   = new_x[m] * zp1;
    float val;
    if (u <= x[0])            val = broad[0];
    else if (u >= x[N_E - 1]) val = broad[N_E - 1];
    else {
        int lo = 0, hi = N_E - 1;
        while (hi - lo > 1) {
            const int mid = (lo + hi) >> 1;
            if (x[mid] <= u) lo = mid; else hi = mid;
        }
        const float w = (u - x[lo]) / (x[hi] - x[lo]);
        val = broad[lo] + w * (broad[hi] - broad[lo]);
    }
    const float ldfac = (float)((1e22 / 3.086e24) * (1e22 / 3.086e24));
    folded[m] = val * zp1 * zp1 * norm[0] * ldfac * resp[m];
}

// ---------------------------------------------------------------------------
// Kernel 5: out = rm @ folded  (328 MB of rm, read once, float4 streamed).
// One block per output channel, LDS tree reduction.
// ---------------------------------------------------------------------------
__global__ void __launch_bounds__(256)
rm_kernel(const float* __restrict__ rm, const float* __restrict__ folded,
          float* __restrict__ out) {
    __shared__ float red[256];
    const int c   = blockIdx.x;
    const int tid = threadIdx.x;
    const float4* __restrict__ row = (const float4*)(rm + (size_t)c * N_NEW);
    const float4* __restrict__ f4  = (const float4*)folded;
    float acc = 0.0f;
    for (int n = tid; n < N_NEW / 4; n += 256) {
        const float4 r = row[n];
        const float4 f = f4[n];
        acc = fmaf(r.x, f.x, acc);
        acc = fmaf(r.y, f.y, acc);
        acc = fmaf(r.z, f.z, acc);
        acc = fmaf(r.w, f.w, acc);
    }
    red[tid] = acc;
    __syncthreads();
    for (int s = 128; s > 0; s >>= 1) {
        if (tid < s) red[tid] += red[tid + s];
        __syncthreads();
    }
    if (tid == 0) out[c] = red[0];
}

// ---------------------------------------------------------------------------
extern "C" void kernel_launch(void* const* d_in, const int* in_sizes, int n_in,
                              void* d_out, int out_size, void* d_ws, size_t ws_size,
                              hipStream_t stream) {
    const float* temp  = (const float*)d_in[0];
    const float* abund = (const float*)d_in[1];
    const float* logz  = (const float*)d_in[2];
    const float* norm  = (const float*)d_in[3];
    const float* vel   = (const float*)d_in[4];
    const float* W1    = (const float*)d_in[5];
    const float* b1    = (const float*)d_in[6];
    const float* W2    = (const float*)d_in[7];
    const float* b2    = (const float*)d_in[8];
    const float* W3    = (const float*)d_in[9];
    const float* b3    = (const float*)d_in[10];
    const float* Wout  = (const float*)d_in[11];
    const float* bout  = (const float*)d_in[12];
    const float* scl   = (const float*)d_in[13];
    const float* mns   = (const float*)d_in[14];
    const float* x     = (const float*)d_in[15];
    const float* dx    = (const float*)d_in[16];
    const float* newx  = (const float*)d_in[17];
    const float* resp  = (const float*)d_in[18];
    const float* rm    = (const float*)d_in[19];
    float* out = (float*)d_out;

    // Workspace layout (all regions written before read; ~1.5 MB total)
    char* ws = (char*)d_ws;
    float* H      = (float*)(ws);                              // 30*152 f32
    float* ypart  = (float*)(ws + 18432);                      // 5 * 50125 f32
    float* y      = (float*)(ws + 18432 + 5 * 200704);         // 50125 f32
    float* broad  = (float*)(ws + 18432 + 6 * 200704);         // 50125 f32
    float* folded = (float*)(ws + 18432 + 7 * 200704);         // 20000 f32

    mlp_kernel<<<NELEM, 160, 0, stream>>>(temp, W1, b1, W2, b2, W3, b3, H);

    dim3 gwout((NGRP32 + 7) / 8, EGRP);                        // (196, 5)
    wout_y_kernel<<<gwout, 256, 0, stream>>>(Wout, bout, scl, mns, abund, H,
                                             ypart);

    combine_y_kernel<<<(N_E + 255) / 256, 256, 0, stream>>>(ypart, y);

    broaden_kernel<<<(N_E + 255) / 256, 256, 0, stream>>>(x, dx, y, vel, broad);

    interp_fold_kernel<<<(N_NEW + 255) / 256, 256, 0, stream>>>(x, broad, newx,
                                                                resp, logz, norm,
                                                                folded);

    rm_kernel<<<N_CHAN, 256, 0, stream>>>(rm, folded, out);
}